// ROIPoolingConv_19250043421473
// MI455X (gfx1250) — compile-verified
//
#include <hip/hip_runtime.h>
#include <stdint.h>

// ROI-Align, MI455X (gfx1250). Memory-bound: ~240 MB HBM traffic -> ~10us floor.
// CDNA5 paths used: global_load_async_to_lds_b128 (ASYNCcnt), s_wait_asynccnt,
// ds_load_b128 readback, non-temporal b128 stores.

typedef float v4f __attribute__((ext_vector_type(4)));
typedef int   v4i __attribute__((ext_vector_type(4)));

#define IMG_H 128
#define IMG_W 128
#define IMG_C 512
#define POOL  14
#define CHUNKS (IMG_C / 4)   // 128 float4 chunks per pixel == blockDim.x

#if defined(__HIP_DEVICE_COMPILE__)
// Builtin signature (from clang diagnostic): (v4i addrspace(1)*, v4i addrspace(3)*, Imm int, Imm int)
typedef __attribute__((address_space(1))) v4i* gv4i_p;
typedef __attribute__((address_space(3))) v4i* lv4i_p;
#define GPTR(p) ((gv4i_p)(p))
#define LPTR(p) ((lv4i_p)(p))
#endif

__global__ __launch_bounds__(CHUNKS) void roi_align_cdna5(
    const float* __restrict__ img,   // (H, W, C) f32
    const int*   __restrict__ rois,  // (NROIS, 4) int32: x, y, w, h
    float*       __restrict__ out)   // (NROIS, POOL, POOL, C) f32
{
    __shared__ v4f tile[4 * CHUNKS];   // 8 KB: 4 neighbor pixels x 128 chunks

    const int t  = threadIdx.x;            // channel chunk 0..127
    const int b  = blockIdx.x;
    const int px = b % POOL;
    const int py = (b / POOL) % POOL;
    const int r  = b / (POOL * POOL);

    const int rx = rois[4 * r + 0];
    const int ry = rois[4 * r + 1];
    const int rw = rois[4 * r + 2];
    const int rh = rois[4 * r + 3];

    // ---- y axis coords (matches reference _axis_coords exactly) ----
    const float hs = (float)rh;
    float sy = ((float)py + 0.5f) * (hs * (1.0f / (float)POOL)) - 0.5f;
    sy = fminf(fmaxf(sy, 0.0f), fmaxf(hs - 1.0f, 0.0f));
    const int   iy0 = (int)floorf(sy);
    const int   iy1 = min(iy0 + 1, rh - 1);
    const float fy  = sy - (float)iy0;
    const int   y0  = ry + iy0;
    const int   y1  = ry + iy1;

    // ---- x axis coords ----
    const float wsz = (float)rw;
    float sx = ((float)px + 0.5f) * (wsz * (1.0f / (float)POOL)) - 0.5f;
    sx = fminf(fmaxf(sx, 0.0f), fmaxf(wsz - 1.0f, 0.0f));
    const int   ix0 = (int)floorf(sx);
    const int   ix1 = min(ix0 + 1, rw - 1);
    const float fx  = sx - (float)ix0;
    const int   x0  = rx + ix0;
    const int   x1  = rx + ix1;

    const v4f* g00 = (const v4f*)(img + (size_t)(y0 * IMG_W + x0) * IMG_C) + t;
    const v4f* g01 = (const v4f*)(img + (size_t)(y0 * IMG_W + x1) * IMG_C) + t;
    const v4f* g10 = (const v4f*)(img + (size_t)(y1 * IMG_W + x0) * IMG_C) + t;
    const v4f* g11 = (const v4f*)(img + (size_t)(y1 * IMG_W + x1) * IMG_C) + t;

    v4f v00, v01, v10, v11;

#if defined(__gfx1250__) && __has_builtin(__builtin_amdgcn_global_load_async_to_lds_b128)
    // CDNA5 async data path: stage 4 neighbor chunks into LDS (ASYNCcnt).
    __builtin_amdgcn_global_load_async_to_lds_b128(GPTR(g00), LPTR(&tile[0 * CHUNKS + t]), 0, 0);
    __builtin_amdgcn_global_load_async_to_lds_b128(GPTR(g01), LPTR(&tile[1 * CHUNKS + t]), 0, 0);
    __builtin_amdgcn_global_load_async_to_lds_b128(GPTR(g10), LPTR(&tile[2 * CHUNKS + t]), 0, 0);
    __builtin_amdgcn_global_load_async_to_lds_b128(GPTR(g11), LPTR(&tile[3 * CHUNKS + t]), 0, 0);
#if __has_builtin(__builtin_amdgcn_s_wait_asynccnt)
    __builtin_amdgcn_s_wait_asynccnt(0);
#else
    asm volatile("s_wait_asynccnt 0" ::: "memory");
#endif
    __syncthreads();
    v00 = tile[0 * CHUNKS + t];
    v01 = tile[1 * CHUNKS + t];
    v10 = tile[2 * CHUNKS + t];
    v11 = tile[3 * CHUNKS + t];
#else
    // Fallback: direct vector loads (still correct, still coalesced).
    v00 = *g00;
    v01 = *g01;
    v10 = *g10;
    v11 = *g11;
#endif

    // Exact algebraic form of the reference: a*(1-f) + b*f
    const v4f top = v00 * (1.0f - fx) + v01 * fx;
    const v4f bot = v10 * (1.0f - fx) + v11 * fx;
    const v4f res = top * (1.0f - fy) + bot * fy;

    v4f* o = (v4f*)(out + (size_t)((r * POOL + py) * POOL + px) * IMG_C) + t;
    // Write-once 205MB stream: non-temporal so the L2 keeps the 33.5MB image.
    __builtin_nontemporal_store(res, o);
}

extern "C" void kernel_launch(void* const* d_in, const int* in_sizes, int n_in,
                              void* d_out, int out_size, void* d_ws, size_t ws_size,
                              hipStream_t stream) {
    const float* img  = (const float*)d_in[0];
    const int*   rois = (const int*)d_in[1];
    float*       out  = (float*)d_out;

    const int n_rois = in_sizes[1] / 4;           // 512
    const int blocks = n_rois * POOL * POOL;      // one block per output pixel

    roi_align_cdna5<<<blocks, CHUNKS, 0, stream>>>(img, rois, out);
}